// SCC_77515569758710
// MI455X (gfx1250) — compile-verified
//
#include <hip/hip_runtime.h>
#include <hip/hip_bf16.h>

typedef float v2f __attribute__((ext_vector_type(2)));
typedef float v8f __attribute__((ext_vector_type(8)));

#define ZERO8 {0.f,0.f,0.f,0.f,0.f,0.f,0.f,0.f}

__device__ __forceinline__ v8f wmma4(v2f a, v2f b, v8f c) {
  // D = A(16x4,f32) * B(4x16,f32) + C(16x16,f32)  -- exact fp32, RNE
  return __builtin_amdgcn_wmma_f32_16x16x4_f32(false, a, false, b, (short)0, c, false, false);
}

// window w in [0,512): b = w>>6, wy = (w>>3)&7, wx = w&7 ; token l in [0,256): r=l>>4, c=l&15
__device__ __forceinline__ size_t qkv_off(int w, int l, int ch) {
  int b = w >> 6, wy = (w >> 3) & 7, wx = w & 7;
  int r = l >> 4, c = l & 15;
  return ((size_t)((b * 128 + wy * 16 + r) * 128 + (wx * 16 + c))) * 192 + ch;
}

// ---------------- 1: per-pixel channel mean/max ----------------
__global__ void k_chanstats(const float* __restrict__ x, float* __restrict__ ca,
                            float* __restrict__ cmx) {
  int wv = threadIdx.x >> 5, lane = threadIdx.x & 31;
  int p = blockIdx.x * 8 + wv;               // 131072 pixels
  const float* row = x + (size_t)p * 192;
  float s = 0.f, m = -3.402823466e38f;
  for (int c = lane; c < 192; c += 32) { float v = row[c]; s += v; m = fmaxf(m, v); }
  for (int off = 16; off; off >>= 1) {
    s += __shfl_xor(s, off, 32);
    m = fmaxf(m, __shfl_xor(m, off, 32));
  }
  if (lane == 0) { ca[p] = s * (1.f / 192.f); cmx[p] = m; }
}

// ---------------- 2: per-(b,c) spatial mean/max ----------------
__global__ void k_spatialstats(const float* __restrict__ x, float* __restrict__ sa,
                               float* __restrict__ sm) {
  int b = blockIdx.x / 192, c = blockIdx.x % 192;
  __shared__ float ss[256], sx[256];
  float s = 0.f, m = -3.402823466e38f;
  const float* base = x + (size_t)b * 16384 * 192 + c;
  for (int p = threadIdx.x; p < 16384; p += 256) {
    float v = base[(size_t)p * 192]; s += v; m = fmaxf(m, v);
  }
  ss[threadIdx.x] = s; sx[threadIdx.x] = m; __syncthreads();
  for (int off = 128; off; off >>= 1) {
    if (threadIdx.x < off) {
      ss[threadIdx.x] += ss[threadIdx.x + off];
      sx[threadIdx.x] = fmaxf(sx[threadIdx.x], sx[threadIdx.x + off]);
    }
    __syncthreads();
  }
  if (threadIdx.x == 0) { sa[blockIdx.x] = ss[0] * (1.f / 16384.f); sm[blockIdx.x] = sx[0]; }
}

// ---------------- 3: SE-style MLPs (192->19->192) ----------------
__global__ void k_se_mlp(const float* __restrict__ sa, const float* __restrict__ sm,
                         const float* l1f_w, const float* l1f_b,
                         const float* l1s_w, const float* l1s_b,
                         const float* l2f_w, const float* l2f_b,
                         const float* l2s_w, const float* l2s_b,
                         float* __restrict__ s1, float* __restrict__ s2) {
  int b = blockIdx.x, t = threadIdx.x;
  __shared__ float h1[19], h2[19];
  if (t < 19) {
    float a = l1f_b[t], c = l2f_b[t];
    for (int j = 0; j < 192; j++) {
      a += l1f_w[t * 192 + j] * sa[b * 192 + j];
      c += l2f_w[t * 192 + j] * sm[b * 192 + j];
    }
    h1[t] = a; h2[t] = c;
  }
  __syncthreads();
  float a = l1s_b[t], c = l2s_b[t];
  for (int j = 0; j < 19; j++) { a += l1s_w[t * 19 + j] * h1[j]; c += l2s_w[t * 19 + j] * h2[j]; }
  s1[b * 192 + t] = a; s2[b * 192 + t] = c;
}

// ---------------- 4: conv3x3 (1->192) x2 + leaky + combine -> qkv ----------------
__global__ void k_qkv(const float* __restrict__ x, const float* __restrict__ ca,
                      const float* __restrict__ cmx,
                      const float* c1w, const float* c1b,
                      const float* c2w, const float* c2b,
                      const float* __restrict__ s1, const float* __restrict__ s2,
                      float* __restrict__ qkv) {
  int pix = blockIdx.x;                   // b*16384 + y*128 + xx
  int b = pix >> 14, y = (pix >> 7) & 127, xx = pix & 127;
  __shared__ float n1[9], n2[9];
  int t = threadIdx.x;
  if (t < 9) {
    int dy = t / 3 - 1, dx = t % 3 - 1;
    int yy = y + dy, xc = xx + dx;
    bool ok = (yy >= 0) && (yy < 128) && (xc >= 0) && (xc < 128);
    int q = (b << 14) + yy * 128 + xc;
    n1[t] = ok ? ca[q] : 0.f;
    n2[t] = ok ? cmx[q] : 0.f;
  }
  __syncthreads();
  float a1 = c1b[t], a2 = c2b[t];
#pragma unroll
  for (int k = 0; k < 9; k++) { a1 += c1w[t * 9 + k] * n1[k]; a2 += c2w[t * 9 + k] * n2[k]; }
  a1 = a1 > 0.f ? a1 : 0.2f * a1;
  a2 = a2 > 0.f ? a2 : 0.2f * a2;
  size_t o = (size_t)pix * 192 + t;
  qkv[o] = 0.5f * (a1 * s1[b * 192 + t] + a2 * s2[b * 192 + t]) + x[o];
}

// ---------------- 5: DynamicPosBias MLP: (961,2)->12->12->12->6 ----------------
__device__ __forceinline__ void ln_relu12(const float* h, float* o, const float* g,
                                          const float* bb) {
  float m = 0.f;
  for (int k = 0; k < 12; k++) m += h[k];
  m *= (1.f / 12.f);
  float v = 0.f;
  for (int k = 0; k < 12; k++) { float d = h[k] - m; v += d * d; }
  v *= (1.f / 12.f);
  float inv = rsqrtf(v + 1e-5f);
  for (int k = 0; k < 12; k++) {
    float tt = (h[k] - m) * inv * g[k] + bb[k];
    o[k] = tt > 0.f ? tt : 0.f;
  }
}

__global__ void k_posmlp(const float* pp_w, const float* pp_b,
                         const float* g1, const float* b1, const float* w1, const float* bb1,
                         const float* g2, const float* b2, const float* w2, const float* bb2,
                         const float* g3, const float* b3, const float* w3, const float* bb3,
                         float* __restrict__ pos) {
  int r = blockIdx.x * 256 + threadIdx.x;
  if (r >= 961) return;
  float f0 = (float)(r / 31 - 15), f1 = (float)(r % 31 - 15);
  float h[12], o[12];
  for (int k = 0; k < 12; k++) h[k] = pp_w[k * 2] * f0 + pp_w[k * 2 + 1] * f1 + pp_b[k];
  ln_relu12(h, o, g1, b1);
  for (int k = 0; k < 12; k++) {
    float a = bb1[k];
    for (int j = 0; j < 12; j++) a += w1[k * 12 + j] * o[j];
    h[k] = a;
  }
  ln_relu12(h, o, g2, b2);
  for (int k = 0; k < 12; k++) {
    float a = bb2[k];
    for (int j = 0; j < 12; j++) a += w2[k * 12 + j] * o[j];
    h[k] = a;
  }
  ln_relu12(h, o, g3, b3);
  for (int n = 0; n < 6; n++) {
    float a = bb3[n];
    for (int j = 0; j < 12; j++) a += w3[n * 12 + j] * o[j];
    pos[r * 6 + n] = a;
  }
}

// ---------------- 6: rpb (NH,L,LB) = mean over 2x2 of pos[relidx] ----------------
__global__ void k_rpb(const float* __restrict__ pos, float* __restrict__ rpb) {
  int t = blockIdx.x * 256 + threadIdx.x;       // n*16384 + l*64 + m
  int n = t >> 14, l = (t >> 6) & 255, m = t & 63;
  int i1 = l >> 4, j1 = l & 15;
  int bw0 = m >> 3, bw1 = m & 7;
  float s = 0.f;
#pragma unroll
  for (int rh = 0; rh < 2; rh++)
#pragma unroll
    for (int rw = 0; rw < 2; rw++) {
      int row = bw0 * 2 + rh, col = bw1 * 2 + rw;
      int idx = (i1 - row + 15) * 31 + (j1 - col + 15);
      s += pos[idx * 6 + n];
    }
  rpb[t] = s * 0.25f;
}

// ---------------- 7: k = (q@Kg1^T + v@Kg2^T + b)/2 via WMMA; 2x2 downsample k,v ----------------
__global__ void k_kgen(const float* __restrict__ qkv,
                       const float* kg1_w, const float* kg1_b,
                       const float* kg2_w, const float* kg2_b,
                       const float* sl_w, const float* sl_b,
                       float* __restrict__ kbuf, float* __restrict__ kP,
                       float* __restrict__ vP) {
  int wn = blockIdx.x;            // w*6+n
  int w = wn / 6, n = wn % 6;
  int wv = threadIdx.x >> 5, lane = threadIdx.x & 31;
  int mrow = lane & 15, khalf = (lane >> 4) * 2, ncol = lane & 15, rowh = 8 * (lane >> 4);
  __shared__ float ks_lds[256 * 16];          // 16 KB: full per-head k window
  for (int mi = 0; mi < 2; mi++) {
    int mt = wv * 2 + mi;
    v8f acc = ZERO8;
#pragma unroll
    for (int ks = 0; ks < 4; ks++) {
      int kb = ks * 4 + khalf;
      const float* qrow = qkv + qkv_off(w, mt * 16 + mrow, 0);
      v2f aq = {qrow[n * 16 + kb], qrow[n * 16 + kb + 1]};
      v2f av = {qrow[96 + n * 16 + kb], qrow[96 + n * 16 + kb + 1]};
      v2f bq = {kg1_w[ncol * 16 + kb], kg1_w[ncol * 16 + kb + 1]};
      v2f bv = {kg2_w[ncol * 16 + kb], kg2_w[ncol * 16 + kb + 1]};
      acc = wmma4(aq, bq, acc);
      acc = wmma4(av, bv, acc);
    }
    float bias = 0.5f * (kg1_b[ncol] + kg2_b[ncol]);
#pragma unroll
    for (int i = 0; i < 8; i++) {
      int l = mt * 16 + i + rowh;
      float val = 0.5f * acc[i] + bias;
      kbuf[((size_t)wn * 256 + l) * 16 + ncol] = val;
      ks_lds[l * 16 + ncol] = val;
    }
  }
  __syncthreads();
  float w0 = sl_w[0], w1 = sl_w[1], w2 = sl_w[2], w3 = sl_w[3], sb = sl_b[0];
  for (int e = threadIdx.x; e < 1024; e += 256) {
    int m = e >> 4, d = e & 15;
    int bw0 = m >> 3, bw1 = m & 7;
    int l00 = (bw0 * 2) * 16 + bw1 * 2;       // r=0:(0,0) r=1:(0,1) r=2:(1,0) r=3:(1,1)
    float kv = w0 * ks_lds[l00 * 16 + d] + w1 * ks_lds[(l00 + 1) * 16 + d] +
               w2 * ks_lds[(l00 + 16) * 16 + d] + w3 * ks_lds[(l00 + 17) * 16 + d] + sb;
    int vch = 96 + n * 16 + d;
    float vv = w0 * qkv[qkv_off(w, l00, vch)] + w1 * qkv[qkv_off(w, l00 + 1, vch)] +
               w2 * qkv[qkv_off(w, l00 + 16, vch)] + w3 * qkv[qkv_off(w, l00 + 17, vch)] + sb;
    kP[((size_t)wn * 64 + m) * 16 + d] = kv;
    vP[((size_t)wn * 64 + m) * 16 + d] = vv;
  }
}

// ---------------- 8: corr = q@kP^T/16 + rpb ; xs = corr@vP (chained WMMA via LDS) ----------------
__global__ void k_attn(const float* __restrict__ qkv, const float* __restrict__ kP,
                       const float* __restrict__ vP, const float* __restrict__ rpb,
                       float* __restrict__ stage) {
  int wn = blockIdx.x;
  int w = wn / 6, n = wn % 6;
  int wv = threadIdx.x >> 5, lane = threadIdx.x & 31;
  int mrow = lane & 15, khalf = (lane >> 4) * 2, ncol = lane & 15, rowh = 8 * (lane >> 4);
  __shared__ float corr_s[256 * 64];           // 64 KB
  const float* kPb = kP + (size_t)wn * 64 * 16;
  const float* vPb = vP + (size_t)wn * 64 * 16;
  for (int mi = 0; mi < 2; mi++) {
    int mt = wv * 2 + mi;
    for (int nt = 0; nt < 4; nt++) {
      v8f acc = ZERO8;
#pragma unroll
      for (int ks = 0; ks < 4; ks++) {
        int kb = ks * 4 + khalf;
        const float* qrow = qkv + qkv_off(w, mt * 16 + mrow, n * 16);
        v2f a = {qrow[kb], qrow[kb + 1]};
        int mp = nt * 16 + ncol;
        v2f b = {kPb[mp * 16 + kb], kPb[mp * 16 + kb + 1]};
        acc = wmma4(a, b, acc);
      }
#pragma unroll
      for (int i = 0; i < 8; i++) {
        int l = mt * 16 + i + rowh;
        int col = nt * 16 + ncol;
        corr_s[l * 64 + col] = acc[i] * (1.f / 16.f) + rpb[n * 16384 + l * 64 + col];
      }
    }
  }
  __syncthreads();
  for (int mi = 0; mi < 2; mi++) {
    int mt = wv * 2 + mi;
    v8f acc = ZERO8;
#pragma unroll
    for (int ks = 0; ks < 16; ks++) {
      int kb = ks * 4 + khalf;
      v2f a = {corr_s[(mt * 16 + mrow) * 64 + kb], corr_s[(mt * 16 + mrow) * 64 + kb + 1]};
      v2f b = {vPb[kb * 16 + ncol], vPb[(kb + 1) * 16 + ncol]};
      acc = wmma4(a, b, acc);
    }
#pragma unroll
    for (int i = 0; i < 8; i++) {
      int l = mt * 16 + i + rowh;
      stage[qkv_off(w, l, n * 16 + ncol)] = acc[i];
    }
  }
}

// ---------------- 9: cmap = qc^T@kc/256 ; xc = vc@cmap^T ----------------
__global__ void k_chancorr(const float* __restrict__ qkv, const float* __restrict__ kbuf,
                           float* __restrict__ stage) {
  int w = blockIdx.x;
  int wv = threadIdx.x >> 5, lane = threadIdx.x & 31;    // 12 waves
  int mrow = lane & 15, khalf = (lane >> 4) * 2, ncol = lane & 15, rowh = 8 * (lane >> 4);
  __shared__ float cm_s[96 * 96];              // 36 KB
  // phase 1: 36 tiles of cmap, 3 per wave; K = 256 tokens
  for (int ti = 0; ti < 3; ti++) {
    int t = wv * 3 + ti, mt = t / 6, nt = t % 6;
    v8f acc = ZERO8;
    const float* kcb = kbuf + ((size_t)(w * 6 + nt)) * 256 * 16;
    for (int ks = 0; ks < 64; ks++) {
      int kb = ks * 4 + khalf;                 // token index
      v2f a = {qkv[qkv_off(w, kb, mt * 16 + mrow)], qkv[qkv_off(w, kb + 1, mt * 16 + mrow)]};
      v2f b = {kcb[kb * 16 + ncol], kcb[(kb + 1) * 16 + ncol]};
      acc = wmma4(a, b, acc);
    }
#pragma unroll
    for (int i = 0; i < 8; i++)
      cm_s[(mt * 16 + i + rowh) * 96 + nt * 16 + ncol] = acc[i] * (1.f / 256.f);
  }
  __syncthreads();
  // phase 2: 96 tiles of xc (256x96), 8 per wave; K = 96 channels
  for (int ti = 0; ti < 8; ti++) {
    int t = wv * 8 + ti, mt = t / 6, nt = t % 6;
    v8f acc = ZERO8;
    const float* vrow = qkv + qkv_off(w, mt * 16 + mrow, 96);
#pragma unroll
    for (int ks = 0; ks < 24; ks++) {
      int kb = ks * 4 + khalf;                 // channel d
      v2f a = {vrow[kb], vrow[kb + 1]};
      v2f b = {cm_s[(nt * 16 + ncol) * 96 + kb], cm_s[(nt * 16 + ncol) * 96 + kb + 1]};
      acc = wmma4(a, b, acc);
    }
#pragma unroll
    for (int i = 0; i < 8; i++) {
      int l = mt * 16 + i + rowh;
      stage[qkv_off(w, l, 96 + nt * 16 + ncol)] = acc[i];
    }
  }
}

// ---------------- 10: out = stage @ proj_w^T + proj_b  (131072 x 192 x 192) ----------------
__global__ void k_proj(const float* __restrict__ stage, const float* __restrict__ pw,
                       const float* __restrict__ pb, float* __restrict__ out) {
  int wv = threadIdx.x >> 5, lane = threadIdx.x & 31;
  int mrow = lane & 15, khalf = (lane >> 4) * 2, ncol = lane & 15, rowh = 8 * (lane >> 4);
  size_t mt = (size_t)blockIdx.x * 8 + wv;     // 8192 M-tiles
  const float* arow = stage + (mt * 16 + mrow) * 192;
  for (int np = 0; np < 6; np++) {
    v8f acc0 = ZERO8, acc1 = ZERO8;
    int n0 = np * 32 + ncol, n1 = n0 + 16;
#pragma unroll 8
    for (int ks = 0; ks < 48; ks++) {
      int kb = ks * 4 + khalf;
      v2f a = {arow[kb], arow[kb + 1]};
      v2f b0 = {pw[n0 * 192 + kb], pw[n0 * 192 + kb + 1]};
      v2f b1 = {pw[n1 * 192 + kb], pw[n1 * 192 + kb + 1]};
      acc0 = wmma4(a, b0, acc0);
      acc1 = wmma4(a, b1, acc1);
    }
    float bb0 = pb[n0], bb1 = pb[n1];
#pragma unroll
    for (int i = 0; i < 8; i++) {
      size_t row = mt * 16 + i + rowh;
      out[row * 192 + n0] = acc0[i] + bb0;
      out[row * 192 + n1] = acc1[i] + bb1;
    }
  }
}

extern "C" void kernel_launch(void* const* d_in, const int* in_sizes, int n_in,
                              void* d_out, int out_size, void* d_ws, size_t ws_size,
                              hipStream_t stream) {
  const float* x      = (const float*)d_in[0];
  const float* c1w    = (const float*)d_in[1];
  const float* c1b    = (const float*)d_in[2];
  const float* c2w    = (const float*)d_in[3];
  const float* c2b    = (const float*)d_in[4];
  const float* l1f_w  = (const float*)d_in[5];
  const float* l1f_b  = (const float*)d_in[6];
  const float* l1s_w  = (const float*)d_in[7];
  const float* l1s_b  = (const float*)d_in[8];
  const float* l2f_w  = (const float*)d_in[9];
  const float* l2f_b  = (const float*)d_in[10];
  const float* l2s_w  = (const float*)d_in[11];
  const float* l2s_b  = (const float*)d_in[12];
  const float* sl_w   = (const float*)d_in[13];
  const float* sl_b   = (const float*)d_in[14];
  const float* kg1_w  = (const float*)d_in[15];
  const float* kg1_b  = (const float*)d_in[16];
  const float* kg2_w  = (const float*)d_in[17];
  const float* kg2_b  = (const float*)d_in[18];
  const float* pp_w   = (const float*)d_in[19];
  const float* pp_b   = (const float*)d_in[20];
  const float* ln1_g  = (const float*)d_in[21];
  const float* ln1_b  = (const float*)d_in[22];
  const float* p1_w   = (const float*)d_in[23];
  const float* p1_b   = (const float*)d_in[24];
  const float* ln2_g  = (const float*)d_in[25];
  const float* ln2_b  = (const float*)d_in[26];
  const float* p2_w   = (const float*)d_in[27];
  const float* p2_b   = (const float*)d_in[28];
  const float* ln3_g  = (const float*)d_in[29];
  const float* ln3_b  = (const float*)d_in[30];
  const float* p3_w   = (const float*)d_in[31];
  const float* p3_b   = (const float*)d_in[32];
  const float* proj_w = (const float*)d_in[33];
  const float* proj_b = (const float*)d_in[34];

  float* ws = (float*)d_ws;
  size_t o = 0;
  float* qkv   = ws + o; o += 25165824;   // B*H*W*192
  float* stage = ws + o; o += 25165824;   // pre-proj concat, (B,H,W,192)
  float* kbuf  = ws + o; o += 12582912;   // (Bw,NH,L,HD)
  float* kP    = ws + o; o += 3145728;    // (Bw,NH,LB,HD)
  float* vP    = ws + o; o += 3145728;
  float* ca    = ws + o; o += 131072;
  float* cmx   = ws + o; o += 131072;
  float* sa    = ws + o; o += 1536;
  float* sm    = ws + o; o += 1536;
  float* s1    = ws + o; o += 1536;
  float* s2    = ws + o; o += 1536;
  float* pos   = ws + o; o += 5766;       // (961,6)
  float* rpbB  = ws + o; o += 98304;      // (6,256,64)

  k_chanstats<<<16384, 256, 0, stream>>>(x, ca, cmx);
  k_spatialstats<<<1536, 256, 0, stream>>>(x, sa, sm);
  k_se_mlp<<<8, 192, 0, stream>>>(sa, sm, l1f_w, l1f_b, l1s_w, l1s_b,
                                  l2f_w, l2f_b, l2s_w, l2s_b, s1, s2);
  k_qkv<<<131072, 192, 0, stream>>>(x, ca, cmx, c1w, c1b, c2w, c2b, s1, s2, qkv);
  k_posmlp<<<4, 256, 0, stream>>>(pp_w, pp_b, ln1_g, ln1_b, p1_w, p1_b,
                                  ln2_g, ln2_b, p2_w, p2_b, ln3_g, ln3_b, p3_w, p3_b, pos);
  k_rpb<<<384, 256, 0, stream>>>(pos, rpbB);
  k_kgen<<<3072, 256, 0, stream>>>(qkv, kg1_w, kg1_b, kg2_w, kg2_b, sl_w, sl_b, kbuf, kP, vP);
  k_attn<<<3072, 256, 0, stream>>>(qkv, kP, vP, rpbB, stage);
  k_chancorr<<<512, 384, 0, stream>>>(qkv, kbuf, stage);
  k_proj<<<1024, 256, 0, stream>>>(stage, proj_w, proj_b, (float*)d_out);
}